// Attention_23330262352482
// MI455X (gfx1250) — compile-verified
//
#include <hip/hip_runtime.h>
#include <hip/hip_bf16.h>

typedef __attribute__((ext_vector_type(2))) float v2f;
typedef __attribute__((ext_vector_type(8))) float v8f;

#define WMMA4(a, b, c) \
    __builtin_amdgcn_wmma_f32_16x16x4_f32(false, (a), false, (b), (short)0, (c), false, false)

// Problem constants (fixed by the reference: B=4, T=4096, C=1024, SIZE=64, NLAND=8)
constexpr int Tn  = 4096;
constexpr int Dn  = 64;
constexpr int Mn  = 8;     // landmarks
constexpr int BHn = 64;    // B*H = 4*16
constexpr float SCALE = 0.125f; // 1/sqrt(64)

// Workspace layout (float offsets)
constexpr size_t OFF_QL   = 0;
constexpr size_t OFF_KL   = OFF_QL   + (size_t)BHn * Mn * Dn;   // 32768
constexpr size_t OFF_K2   = OFF_KL   + (size_t)BHn * Mn * Dn;   // 65536
constexpr size_t OFF_PINV = OFF_K2   + (size_t)BHn * Mn * Mn;   // +4096
constexpr size_t OFF_K3V  = OFF_PINV + (size_t)BHn * Mn * Mn;
constexpr size_t OFF_W    = OFF_K3V  + (size_t)BHn * Mn * Dn;
constexpr size_t OFF_GMAX = OFF_W    + (size_t)BHn * Mn * Dn;
constexpr size_t OFF_P    = OFF_GMAX + 128;                      // scores [BH][8][T]

// ---------------------------------------------------------------------------
// 1) Landmarks: segment means over T/M = 512 tokens.
// grid = BH*M (one block per landmark), block = 256: 4 token-groups x 64 dims.
__global__ __launch_bounds__(256) void k_landmarks(const float* __restrict__ q,
                                                   const float* __restrict__ k,
                                                   float* __restrict__ ws) {
    int bh = blockIdx.x / Mn;
    int m  = blockIdx.x % Mn;
    constexpr int SEG = Tn / Mn; // 512
    int d = threadIdx.x & 63;
    int g = threadIdx.x >> 6;    // 0..3
    const float* Qp = q + ((size_t)bh * Tn + (size_t)m * SEG) * Dn + d;
    const float* Kp = k + ((size_t)bh * Tn + (size_t)m * SEG) * Dn + d;
    float sq = 0.f, sk = 0.f;
    for (int i = g; i < SEG; i += 4) {
        sq += Qp[(size_t)i * Dn];
        sk += Kp[(size_t)i * Dn];
    }
    __shared__ float rq[4][64];
    __shared__ float rk[4][64];
    rq[g][d] = sq; rk[g][d] = sk;
    __syncthreads();
    if (g == 0) {
        float tq = rq[0][d] + rq[1][d] + rq[2][d] + rq[3][d];
        float tk = rk[0][d] + rk[1][d] + rk[2][d] + rk[3][d];
        ws[OFF_QL + ((size_t)bh * Mn + m) * Dn + d] = tq * (1.0f / SEG);
        ws[OFF_KL + ((size_t)bh * Mn + m) * Dn + d] = tk * (1.0f / SEG);
    }
}

// ---------------------------------------------------------------------------
// 2) kernel_2 = softmax(q_l k_l^T * scale) + global max of column sums.
// grid=BH, block=64 (thread = (i,j))
__global__ __launch_bounds__(64) void k_kernel2(float* __restrict__ ws) {
    int bh = blockIdx.x;
    const float* ql = ws + OFF_QL + (size_t)bh * Mn * Dn;
    const float* kl = ws + OFF_KL + (size_t)bh * Mn * Dn;
    float* k2 = ws + OFF_K2 + (size_t)bh * Mn * Mn;
    __shared__ float s2[Mn * Mn];
    __shared__ float cs[Mn];
    int i = threadIdx.x / Mn, j = threadIdx.x % Mn;
    float s = 0.f;
    for (int d = 0; d < Dn; ++d) s += ql[i * Dn + d] * kl[j * Dn + d];
    s *= SCALE;
    // softmax across j: lanes (i*8+j) -> octet-aligned within the wave
    float mx = s;
    mx = fmaxf(mx, __shfl_xor(mx, 1, 32));
    mx = fmaxf(mx, __shfl_xor(mx, 2, 32));
    mx = fmaxf(mx, __shfl_xor(mx, 4, 32));
    float e = __expf(s - mx);
    float sum = e;
    sum += __shfl_xor(sum, 1, 32);
    sum += __shfl_xor(sum, 2, 32);
    sum += __shfl_xor(sum, 4, 32);
    float val = e / sum;
    k2[i * Mn + j] = val;
    s2[i * Mn + j] = val;
    __syncthreads();
    if (threadIdx.x < Mn) {
        float c = 0.f;
        for (int r = 0; r < Mn; ++r) c += s2[r * Mn + threadIdx.x];
        cs[threadIdx.x] = c;
    }
    __syncthreads();
    if (threadIdx.x == 0) {
        float mxc = cs[0];
        for (int jj = 1; jj < Mn; ++jj) mxc = fmaxf(mxc, cs[jj]);
        // positive floats: uint bit pattern is order-preserving
        atomicMax((unsigned int*)(ws + OFF_GMAX), __float_as_uint(mxc));
    }
}

// ---------------------------------------------------------------------------
// 3) Newton-Schulz pinv of 8x8 kernel_2. grid=BH, block=64
__global__ __launch_bounds__(64) void k_pinv(float* __restrict__ ws) {
    int bh = blockIdx.x;
    __shared__ float Km[64], Vm[64], KV[64], T1[64], T2[64];
    int t = threadIdx.x, i = t / 8, j = t % 8;
    float gmax = __uint_as_float(*(const unsigned int*)(ws + OFF_GMAX));
    Km[t] = (ws + OFF_K2 + (size_t)bh * 64)[t];
    __syncthreads();
    Vm[t] = Km[j * 8 + i] / gmax;   // (1/gmax) * K^T
    __syncthreads();
    for (int it = 0; it < 6; ++it) {
        float a = 0.f;
        for (int kk = 0; kk < 8; ++kk) a += Km[i * 8 + kk] * Vm[kk * 8 + j];
        KV[t] = a; __syncthreads();
        T1[t] = (i == j ? 7.f : 0.f) - KV[t]; __syncthreads();
        a = 0.f;
        for (int kk = 0; kk < 8; ++kk) a += KV[i * 8 + kk] * T1[kk * 8 + j];
        T2[t] = a; __syncthreads();
        T1[t] = (i == j ? 15.f : 0.f) - T2[t]; __syncthreads();
        a = 0.f;
        for (int kk = 0; kk < 8; ++kk) a += KV[i * 8 + kk] * T1[kk * 8 + j];
        T2[t] = a; __syncthreads();
        T1[t] = (i == j ? 13.f : 0.f) - T2[t]; __syncthreads();
        a = 0.f;
        for (int kk = 0; kk < 8; ++kk) a += Vm[i * 8 + kk] * T1[kk * 8 + j];
        __syncthreads();
        Vm[t] = 0.25f * a; __syncthreads();
    }
    (ws + OFF_PINV + (size_t)bh * 64)[t] = Vm[t];
}

// ---------------------------------------------------------------------------
// 4) Scores P = scale * q_l @ K^T  via WMMA f32 16x16x4. grid=BH*32, block=256 (8 waves)
// A = q_l zero-padded to 16 rows in LDS -> all fragment loads unconditional.
__global__ __launch_bounds__(256) void k_scores(const float* __restrict__ k,
                                                float* __restrict__ ws) {
    int bh = blockIdx.x / 32;
    __shared__ float sql[16 * Dn];   // rows 8..15 zero
    const float* ql = ws + OFF_QL + (size_t)bh * Mn * Dn;
    for (int idx = threadIdx.x; idx < 16 * Dn; idx += blockDim.x)
        sql[idx] = (idx < Mn * Dn) ? ql[idx] : 0.f;
    __syncthreads();
    int wave = threadIdx.x >> 5, lane = threadIdx.x & 31;
    int ttile = (blockIdx.x & 31) * 8 + wave;     // 0..255
    int t0 = ttile * 16;
    const float* K = k + (size_t)bh * Tn * Dn;
    int n = lane & 15, hi = lane >> 4, koff = hi * 2;
    const float* arow = sql + n * Dn + koff;                  // A: row n (zero-padded)
    const float* brow = K + (size_t)(t0 + n) * Dn + koff;     // B: K^T column n
    v8f c = {0.f, 0.f, 0.f, 0.f, 0.f, 0.f, 0.f, 0.f};
    #pragma unroll
    for (int kc = 0; kc < Dn; kc += 4) {
        v2f a, b;
        a[0] = arow[kc]; a[1] = arow[kc + 1];
        b[0] = brow[kc]; b[1] = brow[kc + 1];
        c = WMMA4(a, b, c);
    }
    if (hi == 0) {  // valid rows m=0..7 live in lanes 0..15, VGPRs 0..7
        float* Prow = ws + OFF_P + (size_t)bh * Mn * Tn + t0 + n;
        #pragma unroll
        for (int r = 0; r < 8; ++r) Prow[(size_t)r * Tn] = c[r] * SCALE;
    }
}

// ---------------------------------------------------------------------------
// 5) Row softmax over T (kernel_3 normalization), in place. grid=BH*8, block=256
__global__ __launch_bounds__(256) void k_rowsoftmax(float* __restrict__ ws) {
    float* row = ws + OFF_P + (size_t)blockIdx.x * Tn;
    __shared__ float red[256];
    int tid = threadIdx.x;
    float mx = -1e30f;
    for (int t = tid; t < Tn; t += 256) mx = fmaxf(mx, row[t]);
    red[tid] = mx; __syncthreads();
    for (int s = 128; s > 0; s >>= 1) { if (tid < s) red[tid] = fmaxf(red[tid], red[tid + s]); __syncthreads(); }
    mx = red[0]; __syncthreads();
    float sum = 0.f;
    for (int t = tid; t < Tn; t += 256) { float e = __expf(row[t] - mx); row[t] = e; sum += e; }
    red[tid] = sum; __syncthreads();
    for (int s = 128; s > 0; s >>= 1) { if (tid < s) red[tid] += red[tid + s]; __syncthreads(); }
    float inv = 1.f / red[0];
    for (int t = tid; t < Tn; t += 256) row[t] *= inv;
}

// ---------------------------------------------------------------------------
// 6) k3v = P @ V via WMMA; T split into 8 segments, combined with atomicAdd.
// tasks = BH(64) * ntile(4) * tseg(8) = 2048 waves -> grid=256, block=256
// A rows 8..15 are zeroed via clamp+mask (v_cndmask, no EXEC divergence).
__global__ __launch_bounds__(256) void k_k3v(const float* __restrict__ v,
                                             float* __restrict__ ws) {
    int wave = threadIdx.x >> 5, lane = threadIdx.x & 31;
    int task = blockIdx.x * 8 + wave;
    int bh = task >> 5;
    int ntile = (task >> 3) & 3;
    int tseg = task & 7;
    const float* V = v + (size_t)bh * Tn * Dn;
    int n = lane & 15, hi = lane >> 4, koff = hi * 2;
    float msk = (n < Mn) ? 1.f : 0.f;
    const float* Prow = ws + OFF_P + ((size_t)bh * Mn + (n & 7)) * Tn; // clamped, always valid
    int d0 = ntile * 16;
    int tbeg = tseg * (Tn / 8);
    const float* vcol = V + (size_t)tbeg * Dn + d0 + n;
    v8f c = {0.f, 0.f, 0.f, 0.f, 0.f, 0.f, 0.f, 0.f};
    for (int tc = 0; tc < Tn / 8; tc += 4) {
        v2f a, b;
        a[0] = Prow[tbeg + tc + koff]     * msk;
        a[1] = Prow[tbeg + tc + koff + 1] * msk;
        b[0] = vcol[(size_t)(tc + koff)     * Dn];
        b[1] = vcol[(size_t)(tc + koff + 1) * Dn];
        c = WMMA4(a, b, c);
    }
    if (hi == 0) {
        float* outp = ws + OFF_K3V + (size_t)bh * Mn * Dn + d0 + n;
        #pragma unroll
        for (int r = 0; r < 8; ++r) atomicAdd(&outp[(size_t)r * Dn], c[r]);
    }
}

// ---------------------------------------------------------------------------
// 7) W = pinv @ k3v  (8x8 @ 8x64). grid=BH, block=64
__global__ __launch_bounds__(64) void k_W(float* __restrict__ ws) {
    int bh = blockIdx.x, d = threadIdx.x;
    __shared__ float pv[64];
    pv[d] = (ws + OFF_PINV + (size_t)bh * 64)[d];
    __syncthreads();
    const float* k3v = ws + OFF_K3V + (size_t)bh * Mn * Dn;
    float* W = ws + OFF_W + (size_t)bh * Mn * Dn;
    for (int i = 0; i < Mn; ++i) {
        float s = 0.f;
        for (int j = 0; j < Mn; ++j) s += pv[i * 8 + j] * k3v[(size_t)j * Dn + d];
        W[(size_t)i * Dn + d] = s;
    }
}

// ---------------------------------------------------------------------------
// 8) out = softmax(Q k_l^T * scale) @ W, fused per 16-row tile. grid=BH*32, block=256
__global__ __launch_bounds__(256) void k_out(const float* __restrict__ q,
                                             float* __restrict__ out,
                                             const float* __restrict__ ws) {
    int bh = blockIdx.x / 32;
    __shared__ float skl[16 * Dn];    // k_l zero-padded to 16 rows
    __shared__ float sW[Mn * Dn];
    __shared__ float sA[8][16][16];   // per-wave C->A relayout buffers
    const float* kl = ws + OFF_KL + (size_t)bh * Mn * Dn;
    const float* Wg = ws + OFF_W + (size_t)bh * Mn * Dn;
    for (int idx = threadIdx.x; idx < 16 * Dn; idx += blockDim.x)
        skl[idx] = (idx < Mn * Dn) ? kl[idx] : 0.f;
    for (int idx = threadIdx.x; idx < Mn * Dn; idx += blockDim.x)
        sW[idx] = Wg[idx];
    __syncthreads();
    int wave = threadIdx.x >> 5, lane = threadIdx.x & 31;
    int ttile = (blockIdx.x & 31) * 8 + wave;
    int t0 = ttile * 16;
    const float* Q = q + (size_t)bh * Tn * Dn;
    int n = lane & 15, hi = lane >> 4, koff = hi * 2;

    // S = Q_tile @ k_l^T  (B cols 8..15 zero-padded in LDS)
    const float* arow = Q + (size_t)(t0 + n) * Dn + koff;
    const float* brow = skl + n * Dn + koff;
    v8f c = {0.f, 0.f, 0.f, 0.f, 0.f, 0.f, 0.f, 0.f};
    #pragma unroll
    for (int kc = 0; kc < Dn; kc += 4) {
        v2f a, b;
        a[0] = arow[kc]; a[1] = arow[kc + 1];
        b[0] = brow[kc]; b[1] = brow[kc + 1];
        c = WMMA4(a, b, c);
    }
    // per-row softmax over the 8 landmark columns (octet shfl reduction fits C layout)
    float (*A)[16] = sA[wave];
    #pragma unroll
    for (int r = 0; r < 8; ++r) {
        float s = c[r] * SCALE;
        float masked = (n < Mn) ? s : -1e30f;
        float mx = masked;
        mx = fmaxf(mx, __shfl_xor(mx, 1, 32));
        mx = fmaxf(mx, __shfl_xor(mx, 2, 32));
        mx = fmaxf(mx, __shfl_xor(mx, 4, 32));
        float e = (n < Mn) ? __expf(s - mx) : 0.f;
        float sum = e;
        sum += __shfl_xor(sum, 1, 32);
        sum += __shfl_xor(sum, 2, 32);
        sum += __shfl_xor(sum, 4, 32);
        int m = r + 8 * hi;
        A[m][n] = (n < Mn) ? (e / sum) : 0.f;
    }
    // wave-local LDS store->load; LDS ops from one wave are in order.
    // out_tile = A(16x8) @ W(8x64)
    float* orow = out + (size_t)bh * Tn * Dn;
    #pragma unroll
    for (int d0 = 0; d0 < Dn; d0 += 16) {
        v8f c2 = {0.f, 0.f, 0.f, 0.f, 0.f, 0.f, 0.f, 0.f};
        #pragma unroll
        for (int kc = 0; kc < Mn; kc += 4) {
            v2f a, b;
            a[0] = A[n][kc + koff];
            a[1] = A[n][kc + koff + 1];
            b[0] = sW[(size_t)(kc + koff)     * Dn + d0 + n];
            b[1] = sW[(size_t)(kc + koff + 1) * Dn + d0 + n];
            c2 = WMMA4(a, b, c2);
        }
        #pragma unroll
        for (int r = 0; r < 8; ++r) {
            int m = r + 8 * hi;
            orow[(size_t)(t0 + m) * Dn + d0 + n] = c2[r];
        }
    }
}

// ---------------------------------------------------------------------------
extern "C" void kernel_launch(void* const* d_in, const int* in_sizes, int n_in,
                              void* d_out, int out_size, void* d_ws, size_t ws_size,
                              hipStream_t stream) {
    const float* q = (const float*)d_in[0];
    const float* k = (const float*)d_in[1];
    const float* v = (const float*)d_in[2];
    float* out = (float*)d_out;
    float* ws = (float*)d_ws;

    // zero the accumulators we need (graph-capturable memset nodes)
    hipMemsetAsync(ws + OFF_GMAX, 0, sizeof(float), stream);
    hipMemsetAsync(ws + OFF_K3V, 0, (size_t)BHn * Mn * Dn * sizeof(float), stream);

    k_landmarks <<<BHn * Mn, 256, 0, stream>>>(q, k, ws);
    k_kernel2   <<<BHn,       64, 0, stream>>>(ws);
    k_pinv      <<<BHn,       64, 0, stream>>>(ws);
    k_scores    <<<BHn * 32, 256, 0, stream>>>(k, ws);
    k_rowsoftmax<<<BHn * Mn, 256, 0, stream>>>(ws);
    k_k3v       <<<256,      256, 0, stream>>>(v, ws);
    k_W         <<<BHn,       64, 0, stream>>>(ws);
    k_out       <<<BHn * 32, 256, 0, stream>>>(q, out, ws);
}